// judge_layer_50096498540623
// MI455X (gfx1250) — compile-verified
//
#include <hip/hip_runtime.h>

// Quadrant one-hot: in = (N,2) f32, out = (N,4) f32.
// q1=(x0>0)&(x1>0), q2=(x0<0)&(x1>0), q3=(x0<0)&(x1<0), q4=(x0>0)&(x1<0)
// Strict compares -> zero coordinates give all-zero rows (matches reference).
//
// Two-kernel structure so the main kernel is 100% branch-free:
//   - main: every thread handles exactly ITER=4 in-range 16B chunks
//           (grid-stride layout). 4 NT b128 loads -> selects -> 8 NT b128
//           stores. No bounds checks, no exec-mask diamonds.
//   - tail: guarded, one chunk/thread; launched only if nchunks is not a
//           multiple of BLOCK*ITER (never, for this problem size).

typedef float v4f __attribute__((ext_vector_type(4)));

static constexpr int BLOCK = 256;  // 8 wave32 waves per workgroup
static constexpr int ITER  = 4;    // independent b128 loads in flight / thread

__device__ __forceinline__ v4f quad(float x0, float x1) {
  const bool p0 = x0 > 0.0f, n0 = x0 < 0.0f;
  const bool p1 = x1 > 0.0f, n1 = x1 < 0.0f;
  v4f o;
  o[0] = (p0 && p1) ? 1.0f : 0.0f;
  o[1] = (n0 && p1) ? 1.0f : 0.0f;
  o[2] = (n0 && n1) ? 1.0f : 0.0f;
  o[3] = (p0 && n1) ? 1.0f : 0.0f;
  return o;
}

// Main kernel: no bounds logic whatsoever. Host guarantees all 4 chunk
// indices (c0, c0+stride, c0+2s, c0+3s) are < nchunks.
__global__ __launch_bounds__(BLOCK)
void judge_layer_50096498540623_main(const v4f* __restrict__ in,
                                     v4f* __restrict__ out,
                                     unsigned stride) {  // total threads
  const unsigned c0 = blockIdx.x * (unsigned)BLOCK + threadIdx.x;
  const unsigned c1 = c0 + stride;
  const unsigned c2 = c0 + 2u * stride;
  const unsigned c3 = c0 + 3u * stride;

  const v4f a = __builtin_nontemporal_load(&in[c0]);
  const v4f b = __builtin_nontemporal_load(&in[c1]);
  const v4f c = __builtin_nontemporal_load(&in[c2]);
  const v4f d = __builtin_nontemporal_load(&in[c3]);

  __builtin_nontemporal_store(quad(a[0], a[1]), &out[2u * c0]);
  __builtin_nontemporal_store(quad(a[2], a[3]), &out[2u * c0 + 1u]);
  __builtin_nontemporal_store(quad(b[0], b[1]), &out[2u * c1]);
  __builtin_nontemporal_store(quad(b[2], b[3]), &out[2u * c1 + 1u]);
  __builtin_nontemporal_store(quad(c[0], c[1]), &out[2u * c2]);
  __builtin_nontemporal_store(quad(c[2], c[3]), &out[2u * c2 + 1u]);
  __builtin_nontemporal_store(quad(d[0], d[1]), &out[2u * c3]);
  __builtin_nontemporal_store(quad(d[2], d[3]), &out[2u * c3 + 1u]);
}

// Tail kernel: one guarded chunk per thread, covers [base, nchunks).
__global__ __launch_bounds__(BLOCK)
void judge_layer_50096498540623_tail(const v4f* __restrict__ in,
                                     v4f* __restrict__ out,
                                     unsigned base,
                                     unsigned nchunks) {
  const unsigned c = base + blockIdx.x * (unsigned)BLOCK + threadIdx.x;
  if (c < nchunks) {
    const v4f x = __builtin_nontemporal_load(&in[c]);
    __builtin_nontemporal_store(quad(x[0], x[1]), &out[2u * c]);
    __builtin_nontemporal_store(quad(x[2], x[3]), &out[2u * c + 1u]);
  }
}

extern "C" void kernel_launch(void* const* d_in, const int* in_sizes, int n_in,
                              void* d_out, int out_size, void* d_ws, size_t ws_size,
                              hipStream_t stream) {
  (void)n_in; (void)out_size; (void)d_ws; (void)ws_size;

  const v4f* in  = (const v4f*)d_in[0];  // (N,2) f32 viewed as N/2 float4 chunks
  v4f*       out = (v4f*)d_out;          // (N,4) f32 viewed as N   float4 rows

  const unsigned nchunks = (unsigned)((unsigned long long)in_sizes[0] / 4ull);

  // Main kernel covers the largest multiple of BLOCK*ITER chunks.
  const unsigned chunks_per_block = (unsigned)(BLOCK * ITER);  // 1024
  const unsigned full_blocks = nchunks / chunks_per_block;
  if (full_blocks > 0) {
    const unsigned stride = full_blocks * (unsigned)BLOCK;  // total threads
    judge_layer_50096498540623_main<<<full_blocks, BLOCK, 0, stream>>>(
        in, out, stride);
  }

  // Remainder (0 for this problem size: 2^23 % 1024 == 0).
  const unsigned base = full_blocks * chunks_per_block;
  const unsigned rem  = nchunks - base;
  if (rem > 0) {
    const unsigned tail_blocks = (rem + BLOCK - 1u) / (unsigned)BLOCK;
    judge_layer_50096498540623_tail<<<tail_blocks, BLOCK, 0, stream>>>(
        in, out, base, nchunks);
  }
}